// LipResConv_4973572129039
// MI455X (gfx1250) — compile-verified
//
#include <hip/hip_runtime.h>
#include <hip/hip_bf16.h>

typedef __attribute__((ext_vector_type(2))) float v2f;
typedef __attribute__((ext_vector_type(8))) float v8f;

#define PI_F 3.14159265358979323846f

__device__ __forceinline__ float2 cmul(float2 a, float2 b) {
  return make_float2(a.x * b.x - a.y * b.y, a.x * b.y + a.y * b.x);
}

// ---------------------------------------------------------------------------
// fp32 WMMA 16x16x4 machinery (CDNA5 V_WMMA_F32_16X16X4_F32), planar complex.
// A frag (16x4): lane m = lane&15, VGPRs hold K = {kh, kh+1}, kh = (lane>>4)*2
// B frag (4x16): lane n = lane&15, VGPRs hold K = {kh, kh+1}
// C/D (16x16):   lane n = lane&15, VGPR r holds row m = r + 8*(lane>>4)
// Planar storage: separate Re / Im float planes -> fragment loads are plain
// b64 / 2addr_b32 LDS loads with zero repacking VALU.
// ---------------------------------------------------------------------------
__device__ __forceinline__ v8f wmma_f32x4(v2f a, v2f b, v8f c) {
  return __builtin_amdgcn_wmma_f32_16x16x4_f32(false, a, false, b, (short)0, c,
                                               false, false);
}

// Fragment of logical A(m,k) at P[(r0+m)*rs + (k0+k)*cs].
__device__ __forceinline__ void load_frag_a(const float* P, int rs, int cs,
                                            int r0, int k0, v2f& v) {
  int lane = threadIdx.x & 31;
  int m = lane & 15;
  int kh = (lane >> 4) << 1;
  int o = (r0 + m) * rs + (k0 + kh) * cs;
  v[0] = P[o];
  v[1] = P[o + cs];
}

// Fragment of logical B(k,n) at P[(k0+k)*rs + (n0+n)*cs].
__device__ __forceinline__ void load_frag_b(const float* P, int rs, int cs,
                                            int k0, int n0, v2f& v) {
  int lane = threadIdx.x & 31;
  int n = lane & 15;
  int kh = (lane >> 4) << 1;
  int o = (k0 + kh) * rs + (n0 + n) * cs;
  v[0] = P[o];
  v[1] = P[o + rs];
}

__device__ __forceinline__ void store_ctile(float* Cre, float* Cim, int ld,
                                            int r0, int n0, v8f cr, v8f ci,
                                            float alpha, bool add) {
  int lane = threadIdx.x & 31;
  int n = lane & 15;
  int mo = (lane >> 4) * 8;
#pragma unroll
  for (int r = 0; r < 8; ++r) {
    size_t o = (size_t)(r0 + mo + r) * ld + (n0 + n);
    float vx = alpha * cr[r];
    float vy = alpha * ci[r];
    if (add) { Cre[o] += vx; Cim[o] += vy; }
    else     { Cre[o] = vx;  Cim[o] = vy; }
  }
}

__device__ __forceinline__ void store_ctile_real(float* C, int ld, int r0,
                                                 int n0, v8f cr, float alpha) {
  int lane = threadIdx.x & 31;
  int n = lane & 15;
  int mo = (lane >> 4) * 8;
#pragma unroll
  for (int r = 0; r < 8; ++r)
    C[(size_t)(r0 + mo + r) * ld + (n0 + n)] = alpha * cr[r];
}

// Block-cooperative complex GEMM: C (+)= alpha * opA(A) @ opB(B).
// Conjugation flags are compile-time; strides express transpose.
// M,N multiples of 16; K multiple of 4. Tile loop wave-uniform -> EXEC all 1s.
template <bool CA, bool CB>
__device__ void cgemm_block(const float* Are, const float* Aim, int a_rs, int a_cs,
                            const float* Bre, const float* Bim, int b_rs, int b_cs,
                            float* Cre, float* Cim, int c_ld,
                            int M, int N, int K, float alpha, bool add) {
  int wave = (int)(threadIdx.x >> 5);
  int nwaves = (int)(blockDim.x >> 5);
  int mt = M >> 4, nt = N >> 4;
  for (int t = wave; t < mt * nt; t += nwaves) {
    int tm = (t / nt) << 4;
    int tn = (t % nt) << 4;
    v8f cr = {}; v8f ci = {};
    for (int k = 0; k < K; k += 4) {
      v2f ar, ai, br, bi;
      load_frag_a(Are, a_rs, a_cs, tm, k, ar);
      load_frag_a(Aim, a_rs, a_cs, tm, k, ai);
      load_frag_b(Bre, b_rs, b_cs, k, tn, br);
      load_frag_b(Bim, b_rs, b_cs, k, tn, bi);
      if (!CA && !CB) {
        v2f nai = -ai;
        cr = wmma_f32x4(ar, br, cr);
        cr = wmma_f32x4(nai, bi, cr);
        ci = wmma_f32x4(ar, bi, ci);
        ci = wmma_f32x4(ai, br, ci);
      } else if (CA && !CB) {
        v2f nai = -ai;
        cr = wmma_f32x4(ar, br, cr);
        cr = wmma_f32x4(ai, bi, cr);
        ci = wmma_f32x4(ar, bi, ci);
        ci = wmma_f32x4(nai, br, ci);
      } else if (!CA && CB) {
        v2f nbi = -bi;
        cr = wmma_f32x4(ar, br, cr);
        cr = wmma_f32x4(ai, bi, cr);
        ci = wmma_f32x4(ar, nbi, ci);
        ci = wmma_f32x4(ai, br, ci);
      } else {
        v2f nai = -ai; v2f nbi = -bi;
        cr = wmma_f32x4(ar, br, cr);
        cr = wmma_f32x4(nai, bi, cr);
        ci = wmma_f32x4(ar, nbi, ci);
        ci = wmma_f32x4(nai, br, ci);
      }
    }
    store_ctile(Cre, Cim, c_ld, tm, tn, cr, ci, alpha, add);
  }
}

// A purely real (imag plane absent): 2 WMMAs per k-step.
__device__ void cgemm_block_rA(const float* A, int a_rs, int a_cs,
                               const float* Bre, const float* Bim, int b_rs, int b_cs,
                               float* Cre, float* Cim, int c_ld,
                               int M, int N, int K, float alpha) {
  int wave = (int)(threadIdx.x >> 5);
  int nwaves = (int)(blockDim.x >> 5);
  int mt = M >> 4, nt = N >> 4;
  for (int t = wave; t < mt * nt; t += nwaves) {
    int tm = (t / nt) << 4;
    int tn = (t % nt) << 4;
    v8f cr = {}; v8f ci = {};
    for (int k = 0; k < K; k += 4) {
      v2f ar, br, bi;
      load_frag_a(A, a_rs, a_cs, tm, k, ar);
      load_frag_b(Bre, b_rs, b_cs, k, tn, br);
      load_frag_b(Bim, b_rs, b_cs, k, tn, bi);
      cr = wmma_f32x4(ar, br, cr);
      ci = wmma_f32x4(ar, bi, ci);
    }
    store_ctile(Cre, Cim, c_ld, tm, tn, cr, ci, alpha, false);
  }
}

// Real part of complex product only: 2 WMMAs per k-step.
__device__ void cgemm_block_realout(const float* Are, const float* Aim, int a_rs, int a_cs,
                                    const float* Bre, const float* Bim, int b_rs, int b_cs,
                                    float* C, int c_ld, int M, int N, int K,
                                    float alpha) {
  int wave = (int)(threadIdx.x >> 5);
  int nwaves = (int)(blockDim.x >> 5);
  int mt = M >> 4, nt = N >> 4;
  for (int t = wave; t < mt * nt; t += nwaves) {
    int tm = (t / nt) << 4;
    int tn = (t % nt) << 4;
    v8f cr = {};
    for (int k = 0; k < K; k += 4) {
      v2f ar, ai, br, bi;
      load_frag_a(Are, a_rs, a_cs, tm, k, ar);
      load_frag_a(Aim, a_rs, a_cs, tm, k, ai);
      load_frag_b(Bre, b_rs, b_cs, k, tn, br);
      load_frag_b(Bim, b_rs, b_cs, k, tn, bi);
      v2f nai = -ai;
      cr = wmma_f32x4(ar, br, cr);
      cr = wmma_f32x4(nai, bi, cr);
    }
    store_ctile_real(C, c_ld, tm, tn, cr, alpha);
  }
}

// In-place Gauss-Jordan inversion of a 64x64 complex matrix (planar, LDS).
__device__ void cinv64(float* Mre, float* Mim, float2* prow, float2* pcol) {
  const int tid = (int)threadIdx.x;
  const int nthr = (int)blockDim.x;
  for (int p = 0; p < 64; ++p) {
    __syncthreads();
    if (tid < 64) {
      prow[tid] = make_float2(Mre[p * 64 + tid], Mim[p * 64 + tid]);
      pcol[tid] = make_float2(Mre[tid * 64 + p], Mim[tid * 64 + p]);
    }
    __syncthreads();
    float2 d = prow[p];
    float inv = 1.0f / (d.x * d.x + d.y * d.y);
    float2 piv = make_float2(d.x * inv, -d.y * inv);
    for (int e = tid; e < 4096; e += nthr) {
      int i = e >> 6, j = e & 63;
      float2 v;
      if (i == p && j == p) v = piv;
      else if (i == p) v = cmul(prow[j], piv);
      else if (j == p) { float2 t = cmul(pcol[i], piv); v = make_float2(-t.x, -t.y); }
      else {
        float2 s = cmul(cmul(pcol[i], piv), prow[j]);
        v = make_float2(Mre[e] - s.x, Mim[e] - s.y);
      }
      Mre[e] = v.x; Mim[e] = v.y;
    }
  }
  __syncthreads();
}

// ---------------------------------------------------------------------------
// K0: DFT twiddle tables (planar). F = e^{-i 2pi jk/64}; G = e^{+i 2pi jk/64};
// E[k2][j] (48x64, rows >=33 zero) = c_k2/4096 * e^{+i 2pi j k2/64}
// ---------------------------------------------------------------------------
__global__ void k_tables(float* Fre, float* Fim, float* Gre, float* Gim,
                         float* Ere, float* Eim) {
  int tid = blockIdx.x * blockDim.x + threadIdx.x;
  int stride = gridDim.x * blockDim.x;
  const float w = 2.0f * PI_F / 64.0f;
  for (int e = tid; e < 4096; e += stride) {
    int j = e >> 6, k = e & 63;
    float s, c;
    __sincosf(w * (float)((j * k) & 63), &s, &c);
    Fre[e] = c; Fim[e] = -s;
    Gre[e] = c; Gim[e] = s;
  }
  for (int e = tid; e < 48 * 64; e += stride) {
    int k2 = e >> 6, j = e & 63;
    if (k2 < 33) {
      float cc = (k2 == 0 || k2 == 32) ? 1.0f : 2.0f;
      float s, c;
      __sincosf(w * (float)((j * k2) & 63), &s, &c);
      Ere[e] = cc * c / 4096.0f;
      Eim[e] = cc * s / 4096.0f;
    } else {
      Ere[e] = 0.0f; Eim[e] = 0.0f;
    }
  }
}

// ---------------------------------------------------------------------------
// K1: pixel-unshuffle + rfft2 of one (b,ci) 64x64 image -> xfft[f] planar
// (block stride 8192 floats: re then im plane of the 128x32 matrix).
// ---------------------------------------------------------------------------
__global__ void k_rfft2_x(const float* __restrict__ x,
                          const float* __restrict__ Fre, const float* __restrict__ Fim,
                          float* __restrict__ xfft) {
  extern __shared__ char smem[];
  float* xt  = (float*)smem;      // 64x64 real
  float* Tre = xt + 4096;         // 64x48
  float* Tim = Tre + 3072;
  float* Xre = Tim + 3072;        // 64x48
  float* Xim = Xre + 3072;
  int blk = blockIdx.x;
  int b = blk >> 7, ci = blk & 127;
  int c = ci >> 2, d1 = (ci >> 1) & 1, d2 = ci & 1;
  const float* xin = x + ((size_t)(b * 32 + c)) * 128 * 128;
  for (int e = threadIdx.x; e < 4096; e += blockDim.x) {
    int i = e >> 6, j = e & 63;
    xt[e] = xin[(2 * i + d1) * 128 + (2 * j + d2)];
  }
  __syncthreads();
  cgemm_block_rA(xt, 64, 1, Fre, Fim, 64, 1, Tre, Tim, 48, 64, 48, 64, 1.f);
  __syncthreads();
  cgemm_block<false, false>(Fre, Fim, 64, 1, Tre, Tim, 48, 1,
                            Xre, Xim, 48, 64, 48, 64, 1.f, false);
  __syncthreads();
  for (int e = threadIdx.x; e < 2112; e += blockDim.x) {
    int k1 = e / 33, k2 = e - k1 * 33;
    size_t o = (size_t)e * 8192 + ci * 32 + b;
    xfft[o] = Xre[k1 * 48 + k2];
    xfft[o + 4096] = Xim[k1 * 48 + k2];
  }
}

// ---------------------------------------------------------------------------
// K2: per-frequency pass 1 (one block per f).
//  Wf (192x64) via 9-tap DFT of W, *shift, conj.  U = rows 0:64, V = 64:192.
//  A = V^H V; M = I+A+(U-U^H); Gauss-Jordan invert; GT = Minv(I-A);
//  Hx = -2 Minv^H (V^H xf); GTHx = GT @ Hx.
// ---------------------------------------------------------------------------
__global__ void k_freq1(const float* __restrict__ W, const float* __restrict__ xfft,
                        float* __restrict__ gt, float* __restrict__ hxf,
                        float* __restrict__ gthxf) {
  extern __shared__ char smem[];
  float* WfR = (float*)smem;      // 192*64
  float* WfI = WfR + 12288;
  float* AbR = WfI + 12288;       // 64*64 (becomes I - A)
  float* AbI = AbR + 4096;
  float* MbR = AbI + 4096;        // 64*64
  float* MbI = MbR + 4096;
  float* GTR = MbI + 4096;        // 64*64
  float* GTI = GTR + 4096;
  float* xfR = GTI + 4096;        // 128*32
  float* xfI = xfR + 4096;
  float* t1R = xfI + 4096;        // 64*32
  float* t1I = t1R + 2048;
  float* hxR = t1I + 2048;        // 64*32
  float* hxI = hxR + 2048;
  float2* prow = (float2*)(hxI + 2048);  // 64
  float2* pcol = prow + 64;              // 64

  int f = blockIdx.x;
  int k1 = f / 33, k2 = f - k1 * 33;
  size_t base8 = (size_t)f * 8192;
  if (threadIdx.x < 64)
    __builtin_prefetch(&xfft[base8 + (size_t)threadIdx.x * 128], 0, 1);

  const float w = 2.0f * PI_F / 64.0f;
  float ss, sc;
  __sincosf(w * (float)((k1 + k2) & 63), &ss, &sc);
  float2 shift = make_float2(sc, -ss);  // e^{-i 2pi (k1+k2)/64}

  for (int e = threadIdx.x; e < 12288; e += blockDim.x) {
    int cio = e >> 6, co = e & 63;
    const float* wp = W + ((size_t)cio * 64 + co) * 9;
    float2 acc = make_float2(0.f, 0.f);
#pragma unroll
    for (int p = 0; p < 3; ++p)
#pragma unroll
      for (int q = 0; q < 3; ++q) {
        float s, c;
        __sincosf(w * (float)(k1 * p + k2 * q), &s, &c);
        float wv = wp[p * 3 + q];
        acc.x += wv * c;
        acc.y -= wv * s;  // e^{-i ang}
      }
    acc.y = -acc.y;       // conj
    float2 v = cmul(shift, acc);
    WfR[e] = v.x; WfI[e] = v.y;
  }
  __syncthreads();

  const float* VR = WfR + 4096;  // V rows start at 64 (ld 64)
  const float* VI = WfI + 4096;
  // A = V^H @ V
  cgemm_block<true, false>(VR, VI, 1, 64, VR, VI, 64, 1,
                           AbR, AbI, 64, 64, 64, 128, 1.f, false);
  __syncthreads();
  // M = I + A + (U - U^H);  Ab <- I - (A + U - U^H)
  for (int e = threadIdx.x; e < 4096; e += blockDim.x) {
    int i = e >> 6, j = e & 63;
    float ax = AbR[e], ay = AbI[e];
    ax += WfR[e] - WfR[j * 64 + i];
    ay += WfI[e] + WfI[j * 64 + i];
    float d = (i == j) ? 1.0f : 0.0f;
    MbR[e] = ax + d; MbI[e] = ay;
    AbR[e] = d - ax; AbI[e] = -ay;
  }
  cinv64(MbR, MbI, prow, pcol);  // Mb <- inv(I + A)
  // GT = Minv @ (I - A)
  cgemm_block<false, false>(MbR, MbI, 64, 1, AbR, AbI, 64, 1,
                            GTR, GTI, 64, 64, 64, 64, 1.f, false);
  __syncthreads();
  // persist GT; stage xf tile
  for (int e = threadIdx.x; e < 4096; e += blockDim.x) {
    gt[base8 + e] = GTR[e];
    gt[base8 + 4096 + e] = GTI[e];
    xfR[e] = xfft[base8 + e];
    xfI[e] = xfft[base8 + 4096 + e];
  }
  __syncthreads();
  // t1 = V^H @ xf
  cgemm_block<true, false>(VR, VI, 1, 64, xfR, xfI, 32, 1,
                           t1R, t1I, 32, 64, 32, 128, 1.f, false);
  __syncthreads();
  // hx = -2 * Minv^H @ t1
  cgemm_block<true, false>(MbR, MbI, 1, 64, t1R, t1I, 32, 1,
                           hxR, hxI, 32, 64, 32, 64, -2.f, false);
  __syncthreads();
  // gthx = GT @ hx (into t1); persist hx
  cgemm_block<false, false>(GTR, GTI, 64, 1, hxR, hxI, 32, 1,
                            t1R, t1I, 32, 64, 32, 64, 1.f, false);
  size_t base4 = (size_t)f * 4096;
  for (int e = threadIdx.x; e < 2048; e += blockDim.x) {
    hxf[base4 + e] = hxR[e];
    hxf[base4 + 2048 + e] = hxI[e];
  }
  __syncthreads();
  for (int e = threadIdx.x; e < 2048; e += blockDim.x) {
    gthxf[base4 + e] = t1R[e];
    gthxf[base4 + 2048 + e] = t1I[e];
  }
}

// ---------------------------------------------------------------------------
// K3: per (b,co) image: irfft2(GTHx_fft) -> affine+ReLU -> rfft2 -> sig_fft.
// ---------------------------------------------------------------------------
__global__ void k_mid(const float* __restrict__ gthxf,
                      const float* __restrict__ Gre, const float* __restrict__ Gim,
                      const float* __restrict__ Ere, const float* __restrict__ Eim,
                      const float* __restrict__ Fre, const float* __restrict__ Fim,
                      const float* __restrict__ lamb, const float* __restrict__ bias,
                      float* __restrict__ sigf) {
  extern __shared__ char smem[];
  float* b1R = (float*)smem;   // 64x48
  float* b1I = b1R + 3072;
  float* b2R = b1I + 3072;     // 64x48
  float* b2I = b2R + 3072;
  float* yb  = b2I + 3072;     // 64x64 real
  int blk = blockIdx.x;
  int b = blk >> 6, co = blk & 63;
  for (int e = threadIdx.x; e < 64 * 48; e += blockDim.x) {
    int r = e / 48, k2 = e - r * 48;
    if (k2 < 33) {
      size_t o = (size_t)(r * 33 + k2) * 4096 + co * 32 + b;
      b1R[e] = gthxf[o];
      b1I[e] = gthxf[o + 2048];
    } else {
      b1R[e] = 0.f; b1I[e] = 0.f;
    }
  }
  __syncthreads();
  cgemm_block<false, false>(Gre, Gim, 64, 1, b1R, b1I, 48, 1,
                            b2R, b2I, 48, 64, 48, 64, 1.f, false);
  __syncthreads();
  cgemm_block_realout(b2R, b2I, 48, 1, Ere, Eim, 64, 1,
                      yb, 64, 64, 64, 48, 1.f);
  __syncthreads();
  float scale = -1.0f / (0.5f + __expf(lamb[co]));  // -L / Lamb, L = 1
  float bb = bias[co];
  for (int e = threadIdx.x; e < 4096; e += blockDim.x) {
    float a = scale * yb[e] + bb;
    yb[e] = a > 0.f ? a : 0.f;
  }
  __syncthreads();
  cgemm_block_rA(yb, 64, 1, Fre, Fim, 64, 1, b1R, b1I, 48, 64, 48, 64, 1.f);
  __syncthreads();
  cgemm_block<false, false>(Fre, Fim, 64, 1, b1R, b1I, 48, 1,
                            b2R, b2I, 48, 64, 48, 64, 1.f, false);
  __syncthreads();
  for (int e = threadIdx.x; e < 2112; e += blockDim.x) {
    int kk1 = e / 33, kk2 = e - kk1 * 33;
    size_t o = (size_t)e * 4096 + co * 32 + b;
    sigf[o] = b2R[kk1 * 48 + kk2];
    sigf[o + 2048] = b2I[kk1 * 48 + kk2];
  }
}

// ---------------------------------------------------------------------------
// K4: per-frequency pass 2: outf = L*Hx_fft + GT^H @ sig_fft (accumulating).
// ---------------------------------------------------------------------------
__global__ void k_freq2(const float* __restrict__ gt, const float* __restrict__ sigf,
                        const float* __restrict__ hxf, float* __restrict__ outf) {
  extern __shared__ char smem[];
  float* GTR = (float*)smem;   // 64x64
  float* GTI = GTR + 4096;
  float* sgR = GTI + 4096;     // 64x32
  float* sgI = sgR + 2048;
  float* hxR = sgI + 2048;     // 64x32
  float* hxI = hxR + 2048;
  int f = blockIdx.x;
  size_t b8 = (size_t)f * 8192, b4 = (size_t)f * 4096;
  for (int e = threadIdx.x; e < 4096; e += blockDim.x) {
    GTR[e] = gt[b8 + e];
    GTI[e] = gt[b8 + 4096 + e];
  }
  for (int e = threadIdx.x; e < 2048; e += blockDim.x) {
    sgR[e] = sigf[b4 + e];
    sgI[e] = sigf[b4 + 2048 + e];
    hxR[e] = hxf[b4 + e];       // L = 1.0
    hxI[e] = hxf[b4 + 2048 + e];
  }
  __syncthreads();
  cgemm_block<true, false>(GTR, GTI, 1, 64, sgR, sgI, 32, 1,
                           hxR, hxI, 32, 64, 32, 64, 1.f, true);
  __syncthreads();
  for (int e = threadIdx.x; e < 2048; e += blockDim.x) {
    outf[b4 + e] = hxR[e];
    outf[b4 + 2048 + e] = hxI[e];
  }
}

// ---------------------------------------------------------------------------
// K5: final irfft2 per (b,co) image -> d_out (real).
// ---------------------------------------------------------------------------
__global__ void k_final(const float* __restrict__ outf,
                        const float* __restrict__ Gre, const float* __restrict__ Gim,
                        const float* __restrict__ Ere, const float* __restrict__ Eim,
                        float* __restrict__ out) {
  extern __shared__ char smem[];
  float* b1R = (float*)smem;   // 64x48
  float* b1I = b1R + 3072;
  float* b2R = b1I + 3072;     // 64x48
  float* b2I = b2R + 3072;
  int blk = blockIdx.x;
  int b = blk >> 6, co = blk & 63;
  for (int e = threadIdx.x; e < 64 * 48; e += blockDim.x) {
    int r = e / 48, k2 = e - r * 48;
    if (k2 < 33) {
      size_t o = (size_t)(r * 33 + k2) * 4096 + co * 32 + b;
      b1R[e] = outf[o];
      b1I[e] = outf[o + 2048];
    } else {
      b1R[e] = 0.f; b1I[e] = 0.f;
    }
  }
  __syncthreads();
  cgemm_block<false, false>(Gre, Gim, 64, 1, b1R, b1I, 48, 1,
                            b2R, b2I, 48, 64, 48, 64, 1.f, false);
  __syncthreads();
  cgemm_block_realout(b2R, b2I, 48, 1, Ere, Eim, 64, 1,
                      out + ((size_t)blk << 12), 64, 64, 64, 48, 1.f);
}

// ---------------------------------------------------------------------------
extern "C" void kernel_launch(void* const* d_in, const int* in_sizes, int n_in,
                              void* d_out, int out_size, void* d_ws, size_t ws_size,
                              hipStream_t stream) {
  (void)in_sizes; (void)n_in; (void)out_size; (void)ws_size;
  const float* x    = (const float*)d_in[0];   // (32,32,128,128)
  const float* W    = (const float*)d_in[1];   // (192,64,3,3)
  const float* lamb = (const float*)d_in[2];   // (64,)
  const float* bias = (const float*)d_in[3];   // (64,)
  float* out = (float*)d_out;                  // (32,64,64,64)

  float* ws = (float*)d_ws;
  size_t off = 0;
  auto alloc = [&](size_t nfloats) {
    float* p = ws + off;
    off += nfloats;
    return p;
  };
  float* F     = alloc(8192);                  // re | im planes
  float* G     = alloc(8192);
  float* E     = alloc(6144);                  // 48x64 re | im
  float* xfft  = alloc((size_t)2112 * 8192);   // per-f: 128x32 re | im
  float* gt    = alloc((size_t)2112 * 8192);   // per-f: 64x64 re | im
  float* hxf   = alloc((size_t)2112 * 4096);   // per-f: 64x32 re | im
  float* gthxf = alloc((size_t)2112 * 4096);
  float* sigf  = alloc((size_t)2112 * 4096);
  float* outfq = alloc((size_t)2112 * 4096);

  const size_t sm_fft   = 16384 * sizeof(float);   // 64 KB
  const size_t sm_freq1 = 65792 * sizeof(float);   // 257 KB (<= 320 KB WGP LDS)
  const size_t sm_mid   = 16384 * sizeof(float);   // 64 KB
  const size_t sm_freq2 = 16384 * sizeof(float);   // 64 KB
  const size_t sm_fin   = 12288 * sizeof(float);   // 48 KB

  k_tables<<<8, 256, 0, stream>>>(F, F + 4096, G, G + 4096, E, E + 3072);
  k_rfft2_x<<<4096, 256, sm_fft, stream>>>(x, F, F + 4096, xfft);
  k_freq1<<<2112, 256, sm_freq1, stream>>>(W, xfft, gt, hxf, gthxf);
  k_mid<<<2048, 256, sm_mid, stream>>>(gthxf, G, G + 4096, E, E + 3072,
                                       F, F + 4096, lamb, bias, sigf);
  k_freq2<<<2112, 256, sm_freq2, stream>>>(gt, sigf, hxf, outfq);
  k_final<<<2048, 256, sm_fin, stream>>>(outfq, G, G + 4096, E, E + 3072, out);
}